// TorsionalEnergyNetwork_58445914964447
// MI455X (gfx1250) — compile-verified
//
#include <hip/hip_runtime.h>
#include <hip/hip_bf16.h>
#include <math.h>

typedef __attribute__((ext_vector_type(2))) float v2f;
typedef __attribute__((ext_vector_type(8))) float v8f;

#define FK_ITERS 20

// ---------------------------------------------------------------------------
// Build local 3x4 affine transforms from dofs[:,0:3]; also copy parent array
// into the ping-pong pointer buffer.
// Row-major 12 floats per node: [r00 r01 r02 tx  r10 r11 r12 ty  r20 r21 r22 tz]
// ---------------------------------------------------------------------------
__device__ __forceinline__ void make_local(float phi, float th, float d, float* r) {
    float sp, cp, st, ct;
    sincosf(phi, &sp, &cp);
    sincosf(th, &st, &ct);
    r[0] = cp * ct;  r[1] = -sp;  r[2]  = cp * st;  r[3]  = d * cp * ct;
    r[4] = sp * ct;  r[5] = cp;   r[6]  = sp * st;  r[7]  = d * sp * ct;
    r[8] = -st;      r[9] = 0.f;  r[10] = ct;       r[11] = -d * st;
}

__global__ void k_build(const float* __restrict__ dofs, float* __restrict__ T,
                        const int* __restrict__ parent, int* __restrict__ p0, int N) {
    int n = blockIdx.x * blockDim.x + threadIdx.x;
    if (n >= N) return;
    make_local(dofs[(size_t)n * 9 + 0], dofs[(size_t)n * 9 + 1], dofs[(size_t)n * 9 + 2],
               T + (size_t)n * 12);
    p0[n] = parent[n];
}

__global__ void k_mask(const float* __restrict__ mdofs, const int* __restrict__ midx,
                       float* __restrict__ T, int K) {
    int i = blockIdx.x * blockDim.x + threadIdx.x;
    if (i >= K) return;
    int node = midx[i];
    make_local(mdofs[(size_t)i * 9 + 0], mdofs[(size_t)i * 9 + 1], mdofs[(size_t)i * 9 + 2],
               T + (size_t)node * 12);
}

__global__ void k_ident(float* __restrict__ T) {
    int i = threadIdx.x;
    if (i < 12) T[i] = (i == 0 || i == 5 || i == 10) ? 1.f : 0.f;
}

// ---------------------------------------------------------------------------
// One pointer-doubling round. Each wave composes 4 nodes with a single
// V_WMMA_F32_16X16X4_F32:
//   A (16x4)  = 4 parent matrices stacked vertically  (rows = 4i+r)
//   B (4x16)  = 4 child  matrices side by side        (cols = 4i+c)
//   D (16x16) diagonal 4x4 blocks = parent_i x child_i
// A layout (ISA 7.12.2): lane m (0-15) holds row m, K=0,1 in v[0],v[1];
// lane m+16 holds row m, K=2,3  -> one contiguous float2 load per lane.
// Implicit bottom row [0 0 0 1] is patched in after an (in-bounds, padded)
// load so EXEC stays all-ones for the WMMA.
// ---------------------------------------------------------------------------
__global__ void k_compose(const float* __restrict__ Tin, float* __restrict__ Tout,
                          const int* __restrict__ pin, int* __restrict__ pout, int N) {
    int lane = threadIdx.x & 31;
    int wave = threadIdx.x >> 5;
    int nGroups = (N + 3) >> 2;
    long g = (long)blockIdx.x * (blockDim.x >> 5) + wave;
    if (g >= nGroups) return;                 // uniform per wave

    int half = lane >> 4;                     // 0: K/M-low half, 1: K/M-high half
    int L    = lane & 15;
    int i    = L >> 2;                        // which 4x4 block (0..3)
    int r    = L & 3;                         // row (A) / column (B,D) index
    int node = (int)(g * 4) + i;
    int nc   = node < N ? node : N - 1;
    int pp   = pin[nc];                       // parent index (gather)

    // A operand: parent row r, columns 2*half .. 2*half+1 (contiguous)
    v2f a;
    {
        const float* pa = Tin + (size_t)pp * 12 + (size_t)r * 4 + 2 * half;
        a.x = pa[0];
        a.y = pa[1];                          // padded buffer makes r==3 load safe
        if (r == 3) { a.x = 0.f; a.y = half ? 1.f : 0.f; }
    }
    // B operand: child column c=r; rows (2*half, 2*half+1)
    v2f b;
    {
        const float* pb = Tin + (size_t)nc * 12;
        b.x = pb[(size_t)(2 * half) * 4 + r];                       // row 0 or 2
        b.y = half ? ((r == 3) ? 1.f : 0.f) : pb[4 + r];            // row 1 or [0001]
    }

    v8f acc = {0.f, 0.f, 0.f, 0.f, 0.f, 0.f, 0.f, 0.f};
    acc = __builtin_amdgcn_wmma_f32_16x16x4_f32(
        /*neg_a=*/false, a, /*neg_b=*/false, b,
        /*c_mod=*/(short)0, acc, /*reuse_a=*/false, /*reuse_b=*/false);

    // Scatter diagonal blocks back: lane holds D[:,L] (half=0 -> M 0..7,
    // half=1 -> M 8..15). Diagonal block of column L lives in this lane iff
    // (half==0 && L<8) || (half==1 && L>=8). db selects the 4-row group.
    bool writer = half ? (L >= 8) : (L < 8);
    if (writer && node < N) {
        int db = (half ? (i - 2) : i) * 4;    // 0 or 4
        bool hi = (db != 0);
        float e0 = hi ? acc[4] : acc[0];
        float e1 = hi ? acc[5] : acc[1];
        float e2 = hi ? acc[6] : acc[2];
        float* ro = Tout + (size_t)node * 12 + r;   // column r of rows 0..2
        ro[0] = e0;
        ro[4] = e1;
        ro[8] = e2;
    }
    // pointer doubling: lanes (half==0, r==0) cover i = 0..3
    if (half == 0 && r == 0 && node < N) pout[node] = pin[pp];
}

// ---------------------------------------------------------------------------
// Torsion energy: grid-stride over M dihedrals, deterministic two-stage
// reduction (per-block partial -> fixed-order final tree).
// ---------------------------------------------------------------------------
__global__ void k_torsion(const float* __restrict__ T, const int* __restrict__ atoms,
                          const float* __restrict__ ktor, const float* __restrict__ nper,
                          const float* __restrict__ delta, float* __restrict__ partials,
                          int M) {
    float acc = 0.f;
    int stride = gridDim.x * blockDim.x;
    for (int m = blockIdx.x * blockDim.x + threadIdx.x; m < M; m += stride) {
        int a0 = atoms[4 * m + 0], a1 = atoms[4 * m + 1];
        int a2 = atoms[4 * m + 2], a3 = atoms[4 * m + 3];
        const float* t0 = T + (size_t)a0 * 12;
        const float* t1 = T + (size_t)a1 * 12;
        const float* t2 = T + (size_t)a2 * 12;
        const float* t3 = T + (size_t)a3 * 12;
        float p1x = t0[3], p1y = t0[7], p1z = t0[11];
        float p2x = t1[3], p2y = t1[7], p2z = t1[11];
        float p3x = t2[3], p3y = t2[7], p3z = t2[11];
        float p4x = t3[3], p4y = t3[7], p4z = t3[11];
        float b1x = p2x - p1x, b1y = p2y - p1y, b1z = p2z - p1z;
        float b2x = p3x - p2x, b2y = p3y - p2y, b2z = p3z - p2z;
        float b3x = p4x - p3x, b3y = p4y - p3y, b3z = p4z - p3z;
        float n1x = b1y * b2z - b1z * b2y;
        float n1y = b1z * b2x - b1x * b2z;
        float n1z = b1x * b2y - b1y * b2x;
        float n2x = b2y * b3z - b2z * b3y;
        float n2y = b2z * b3x - b2x * b3z;
        float n2z = b2x * b3y - b2y * b3x;
        float inb2 = 1.f / (sqrtf(b2x * b2x + b2y * b2y + b2z * b2z) + 1e-8f);
        float cx = n1y * n2z - n1z * n2y;
        float cy = n1z * n2x - n1x * n2z;
        float cz = n1x * n2y - n1y * n2x;
        float s  = (cx * b2x + cy * b2y + cz * b2z) * inb2;
        float c  = n1x * n2x + n1y * n2y + n1z * n2z;
        float chi = atan2f(s, c);
        acc += ktor[m] * (1.f + cosf(nper[m] * chi - delta[m]));
    }
    // wave32 reduction
    for (int off = 16; off > 0; off >>= 1) acc += __shfl_down(acc, off, 32);
    __shared__ float sred[8];
    int lane = threadIdx.x & 31, wid = threadIdx.x >> 5;
    if (lane == 0) sred[wid] = acc;
    __syncthreads();
    if (wid == 0) {
        float v = (lane < (int)(blockDim.x >> 5)) ? sred[lane] : 0.f;
        for (int off = 4; off > 0; off >>= 1) v += __shfl_down(v, off, 32);
        if (lane == 0) partials[blockIdx.x] = v;
    }
}

__global__ void k_final(const float* __restrict__ partials, int n, float* __restrict__ out) {
    __shared__ float s[256];
    float acc = 0.f;
    for (int i = threadIdx.x; i < n; i += blockDim.x) acc += partials[i];
    s[threadIdx.x] = acc;
    __syncthreads();
    for (int off = 128; off > 0; off >>= 1) {
        if ((int)threadIdx.x < off) s[threadIdx.x] += s[threadIdx.x + off];
        __syncthreads();
    }
    if (threadIdx.x == 0) out[0] = s[0];
}

// ---------------------------------------------------------------------------
extern "C" void kernel_launch(void* const* d_in, const int* in_sizes, int n_in,
                              void* d_out, int out_size, void* d_ws, size_t ws_size,
                              hipStream_t stream) {
    const float* masked   = (const float*)d_in[0];
    const float* base     = (const float*)d_in[1];
    const float* ktor     = (const float*)d_in[2];
    const float* nper     = (const float*)d_in[3];
    const float* delta    = (const float*)d_in[4];
    const int*   mask_idx = (const int*)d_in[5];
    const int*   parent   = (const int*)d_in[6];
    const int*   atoms    = (const int*)d_in[7];

    const int M = in_sizes[2];
    const int K = in_sizes[5];
    const int N = in_sizes[6];

    // ws layout: Ta | Tb | pa | pb | partials   (~104 MB total, fits L2)
    const size_t tElems = (size_t)N * 12 + 16;          // +16 float pad for row-3 loads
    float* Ta = (float*)d_ws;
    float* Tb = Ta + tElems;
    int*   pa = (int*)(Tb + tElems);
    int*   pb = pa + N;
    float* partials = (float*)(pb + N);

    k_build<<<(N + 255) / 256, 256, 0, stream>>>(base, Ta, parent, pa, N);
    k_mask <<<(K + 255) / 256, 256, 0, stream>>>(masked, mask_idx, Ta, K);
    k_ident<<<1, 32, 0, stream>>>(Ta);

    const int nGroups = (N + 3) / 4;
    const int blocks  = (nGroups + 7) / 8;              // 8 waves / block (256 thr)
    const float* Tin = Ta; float* Tout = Tb;
    const int*   pin = pa; int*   pout = pb;
    for (int it = 0; it < FK_ITERS; ++it) {
        k_compose<<<blocks, 256, 0, stream>>>(Tin, Tout, pin, pout, N);
        const float* tf = Tin; Tin = Tout; Tout = (float*)tf;
        const int*   tp = pin; pin = pout; pout = (int*)tp;
    }

    const int NB = 1024;
    k_torsion<<<NB, 256, 0, stream>>>(Tin, atoms, ktor, nper, delta, partials, M);
    k_final  <<<1, 256, 0, stream>>>(partials, NB, (float*)d_out);
}